// FrequencyDeformableAttentionBandwidth_60189671686363
// MI455X (gfx1250) — compile-verified
//
#include <hip/hip_runtime.h>
#include <math.h>

// ---------------------------------------------------------------------------
// FrequencyDeformableAttentionBandwidth — full pipeline for MI455X (gfx1250)
//
// Big 1x1 convs (256->32 reduce, 32->256 expand+gate) use f32 WMMA
// (V_WMMA_F32_16X16X4_F32). Middle of pipeline is L2-resident 35x35 work.
// ---------------------------------------------------------------------------

typedef __attribute__((ext_vector_type(2))) float v2f;
typedef __attribute__((ext_vector_type(8))) float v8f;

constexpr int B_   = 4;
constexpr int CIN  = 256;
constexpr int Hh   = 224, Ww = 224;
constexpr int CR   = 32;
constexpr int HW   = Hh * Ww;          // 50176
constexpr int N1   = B_ * HW;          // 200704
constexpr int H1   = 111, W1 = 111;
constexpr int HW1  = H1 * W1;          // 12321
constexpr int HP   = 35;
constexpr int HWP  = HP * HP;          // 1225
constexpr int SH   = 17;               // 35/2 (fftshift roll)
constexpr float PI_F = 3.14159265358979323846f;

static __device__ __forceinline__ int clampi(int v, int lo, int hi) {
  return v < lo ? lo : (v > hi ? hi : v);
}

// ---------------------------------------------------------------------------
// K1: c0 = 1x1 conv 256->32 over x.  GEMM: M=32 (o), K=256 (c), N=B*H*W.
// Workgroup = 8 waves covering 32(M) x 64(N); wave computes one 16x16 tile.
// Grid = N1/64 = 3136 (exact), so EXEC is all-ones everywhere.
// ---------------------------------------------------------------------------
__global__ __launch_bounds__(256) void k_conv1x1_reduce(
    const float* __restrict__ x, const float* __restrict__ w1,
    const float* __restrict__ b1, float* __restrict__ c0) {
  const int lane  = threadIdx.x & 31;
  const int wave  = threadIdx.x >> 5;
  const int mTile = (wave & 1) * 16;
  const int n0    = blockIdx.x * 64 + (wave >> 1) * 16;
  const int bIdx  = n0 / HW;            // 64 | HW, tile never crosses batch
  const int hw0   = n0 - bIdx * HW;
  const int col   = lane & 15;          // B/C/D: N striped across lanes
  const int row   = lane & 15;          // A: M striped across lanes
  const int kk    = (lane < 16) ? 0 : 2;// upper halfwave holds K+2 / M+8

  v8f acc = {};
  const float* wA = w1 + (size_t)(mTile + row) * CIN;
  const float* xB = x + (size_t)bIdx * CIN * HW + hw0 + col;
  for (int k = 0; k < CIN; k += 4) {
    v2f a, b;
    a.x = wA[k + kk];
    a.y = wA[k + kk + 1];
    b.x = xB[(size_t)(k + kk) * HW];
    b.y = xB[(size_t)(k + kk + 1) * HW];
    acc = __builtin_amdgcn_wmma_f32_16x16x4_f32(false, a, false, b,
                                                (short)0, acc, false, false);
  }
  const int rowBase = (lane < 16) ? 0 : 8;
  float* outp = c0 + ((size_t)bIdx * CR + mTile + rowBase) * HW + hw0 + col;
#pragma unroll
  for (int r = 0; r < 8; ++r)
    outp[(size_t)r * HW] = acc[r] + b1[mTile + rowBase + r];
}

// ---------------------------------------------------------------------------
// K2: c1 = 3x3 conv stride 2 pad 0 (32->32): [4,32,224,224] -> [4,32,111,111]
// ---------------------------------------------------------------------------
__global__ __launch_bounds__(256) void k_conv3x3_s2(
    const float* __restrict__ c0, const float* __restrict__ wconv,
    const float* __restrict__ bs, float* __restrict__ c1) {
  int idx = blockIdx.x * blockDim.x + threadIdx.x;
  const int total = B_ * CR * HW1;
  if (idx >= total) return;
  int xo = idx % W1; int t = idx / W1;
  int yo = t % H1;   t /= H1;
  int o  = t % CR;   int b = t / CR;
  float acc = bs[o];
  const float* xin = c0 + (size_t)b * CR * HW + (yo * 2) * Ww + (xo * 2);
  const float* wp  = wconv + (size_t)o * CR * 9;
  for (int c = 0; c < CR; ++c) {
    const float* xc = xin + (size_t)c * HW;
    const float* wc = wp + c * 9;
#pragma unroll
    for (int ky = 0; ky < 3; ++ky)
#pragma unroll
      for (int kx = 0; kx < 3; ++kx)
        acc = fmaf(xc[ky * Ww + kx], wc[ky * 3 + kx], acc);
  }
  c1[idx] = acc;
}

// ---------------------------------------------------------------------------
// K3: 7x7 stride-3 VALID maxpool: [4,32,111,111] -> [4,32,35,35]
// ---------------------------------------------------------------------------
__global__ __launch_bounds__(256) void k_maxpool(
    const float* __restrict__ c1, float* __restrict__ p) {
  int idx = blockIdx.x * blockDim.x + threadIdx.x;
  const int total = B_ * CR * HWP;
  if (idx >= total) return;
  int xo = idx % HP; int t = idx / HP;
  int yo = t % HP;   int bc = t / HP;
  const float* src = c1 + (size_t)bc * HW1 + (yo * 3) * W1 + (xo * 3);
  float m = -__builtin_inff();
  for (int ky = 0; ky < 7; ++ky)
#pragma unroll
    for (int kx = 0; kx < 7; ++kx)
      m = fmaxf(m, src[ky * W1 + kx]);
  p[idx] = m;
}

// ---------------------------------------------------------------------------
// K4: forward 35x35 DFT per (b,c) image; fftshift + band-high mask folded in.
// hb = |F|*mask (shifted layout), phase = angle(F) (shifted layout).
// ---------------------------------------------------------------------------
__global__ __launch_bounds__(256) void k_fft_fwd(
    const float* __restrict__ p, float* __restrict__ hb,
    float* __restrict__ phase) {
  __shared__ float sIn[HWP];
  __shared__ float sRe[HWP];
  __shared__ float sIm[HWP];
  const int img = blockIdx.x;  // b*CR + c
  const float* src = p + (size_t)img * HWP;
  for (int i = threadIdx.x; i < HWP; i += blockDim.x) sIn[i] = src[i];
  __syncthreads();
  const float w0 = -2.0f * PI_F / 35.0f;
  // row pass: for each (row r, freq kx)
  for (int i = threadIdx.x; i < HWP; i += blockDim.x) {
    int kx = i % HP, r = i / HP;
    float re = 0.f, im = 0.f;
    for (int xx = 0; xx < HP; ++xx) {
      float s, c;
      __sincosf(w0 * (float)((kx * xx) % HP), &s, &c);
      float v = sIn[r * HP + xx];
      re = fmaf(v, c, re);
      im = fmaf(v, s, im);
    }
    sRe[i] = re;
    sIm[i] = im;
  }
  __syncthreads();
  // column pass + shift + mag/phase/mask
  for (int i = threadIdx.x; i < HWP; i += blockDim.x) {
    int kx = i % HP, ky = i / HP;
    float re = 0.f, im = 0.f;
    for (int r = 0; r < HP; ++r) {
      float s, c;
      __sincosf(w0 * (float)((ky * r) % HP), &s, &c);
      float gr = sRe[r * HP + kx], gi = sIm[r * HP + kx];
      re += gr * c - gi * s;
      im += gr * s + gi * c;
    }
    int sy = (ky + SH) % HP, sx = (kx + SH) % HP;   // fftshift (roll by 17)
    int dy = sy - SH, dx = sx - SH;
    float mask = (dy * dy + dx * dx > 4) ? 1.0f : 0.0f;  // d > 35//16 == 2
    float mag = sqrtf(re * re + im * im);
    int di = sy * HP + sx;
    hb[(size_t)img * HWP + di]    = mag * mask;
    phase[(size_t)img * HWP + di] = atan2f(im, re);
  }
}

// ---------------------------------------------------------------------------
// K5: offset conv 3x3 pad 1 (32 -> 18 = K2*2 channels) on current hb
// ---------------------------------------------------------------------------
__global__ __launch_bounds__(256) void k_off_conv(
    const float* __restrict__ hbc, const float* __restrict__ woff,
    const float* __restrict__ boff, float* __restrict__ off) {
  int idx = blockIdx.x * blockDim.x + threadIdx.x;
  const int total = B_ * 18 * HWP;
  if (idx >= total) return;
  int xo = idx % HP; int t = idx / HP;
  int yo = t % HP;   t /= HP;
  int oc = t % 18;   int b = t / 18;
  float acc = boff[oc];
  const float* wp = woff + (size_t)oc * CR * 9;
  const float* xb = hbc + (size_t)b * CR * HWP;
  for (int c = 0; c < CR; ++c) {
    const float* xc = xb + (size_t)c * HWP;
    const float* wc = wp + c * 9;
#pragma unroll
    for (int ky = 0; ky < 3; ++ky) {
      int yi = yo + ky - 1;
      if (yi < 0 || yi >= HP) continue;
#pragma unroll
      for (int kx = 0; kx < 3; ++kx) {
        int xi = xo + kx - 1;
        if (xi < 0 || xi >= HP) continue;
        acc = fmaf(xc[yi * HP + xi], wc[ky * 3 + kx], acc);
      }
    }
  }
  off[idx] = acc;
}

// ---------------------------------------------------------------------------
// K6: deformable sample + einsum + ReLU. One thread per (b,y,x), holding all
// 32 output-channel accumulators so each bilinear sample is reused 32x.
// ---------------------------------------------------------------------------
__global__ __launch_bounds__(256) void k_deform(
    const float* __restrict__ hbc, const float* __restrict__ off,
    const float* __restrict__ wreg, const float* __restrict__ breg,
    float* __restrict__ outp) {
  int idx = blockIdx.x * blockDim.x + threadIdx.x;
  const int total = B_ * HWP;
  if (idx >= total) return;
  int xo = idx % HP; int t = idx / HP;
  int yo = t % HP;   int b = t / HP;
  float acc[CR];
#pragma unroll
  for (int o = 0; o < CR; ++o) acc[o] = breg[o];
  const float* xb = hbc + (size_t)b * CR * HWP;
  const float* ob = off + (size_t)b * 18 * HWP + yo * HP + xo;
  for (int k = 0; k < 9; ++k) {
    float dy = ob[(size_t)(2 * k) * HWP];
    float dx = ob[(size_t)(2 * k + 1) * HWP];
    float ys = (float)(yo + k / 3 - 1) + dy;
    float xs = (float)(xo + k % 3 - 1) + dx;
    float y0f = floorf(ys), x0f = floorf(xs);
    float fy = ys - y0f, fx = xs - x0f;
    int y0 = (int)y0f, x0 = (int)x0f;
    int y1 = y0 + 1,   x1 = x0 + 1;
    float v00 = ((y0 >= 0) && (y0 < HP) && (x0 >= 0) && (x0 < HP)) ? 1.f : 0.f;
    float v01 = ((y0 >= 0) && (y0 < HP) && (x1 >= 0) && (x1 < HP)) ? 1.f : 0.f;
    float v10 = ((y1 >= 0) && (y1 < HP) && (x0 >= 0) && (x0 < HP)) ? 1.f : 0.f;
    float v11 = ((y1 >= 0) && (y1 < HP) && (x1 >= 0) && (x1 < HP)) ? 1.f : 0.f;
    int cy0 = clampi(y0, 0, HP - 1), cy1 = clampi(y1, 0, HP - 1);
    int cx0 = clampi(x0, 0, HP - 1), cx1 = clampi(x1, 0, HP - 1);
    float w00 = (1.f - fy) * (1.f - fx) * v00;
    float w01 = (1.f - fy) * fx * v01;
    float w10 = fy * (1.f - fx) * v10;
    float w11 = fy * fx * v11;
    int i00 = cy0 * HP + cx0, i01 = cy0 * HP + cx1;
    int i10 = cy1 * HP + cx0, i11 = cy1 * HP + cx1;
    for (int c = 0; c < CR; ++c) {
      const float* xc = xb + (size_t)c * HWP;
      float s = w00 * xc[i00] + w01 * xc[i01] + w10 * xc[i10] + w11 * xc[i11];
      const float* wp = wreg + (size_t)c * 9 + k;  // stride CR*9 per o
#pragma unroll
      for (int o = 0; o < CR; ++o)
        acc[o] = fmaf(wp[(size_t)o * CR * 9], s, acc[o]);
    }
  }
  float* op = outp + (size_t)b * CR * HWP + yo * HP + xo;
#pragma unroll
  for (int o = 0; o < CR; ++o)
    op[(size_t)o * HWP] = fmaxf(acc[o], 0.0f);
}

// ---------------------------------------------------------------------------
// K7: rec = Re(ifft2(ifftshift(hb * e^{i*phase}))). ifftshift folded into
// the spectrum load; naive 35-pt inverse DFT (rows then cols) in LDS.
// ---------------------------------------------------------------------------
__global__ __launch_bounds__(256) void k_fft_inv(
    const float* __restrict__ hb, const float* __restrict__ phase,
    float* __restrict__ rec) {
  __shared__ float aRe[HWP];
  __shared__ float aIm[HWP];
  __shared__ float gRe[HWP];
  __shared__ float gIm[HWP];
  const int img = blockIdx.x;
  const float* hp = hb + (size_t)img * HWP;
  const float* pp = phase + (size_t)img * HWP;
  for (int i = threadIdx.x; i < HWP; i += blockDim.x) {
    int kx = i % HP, ky = i / HP;
    int si = ((ky + SH) % HP) * HP + ((kx + SH) % HP);  // ifftshift gather
    float m = hp[si];
    float s, c;
    __sincosf(pp[si], &s, &c);
    aRe[i] = m * c;
    aIm[i] = m * s;
  }
  __syncthreads();
  const float w0 = 2.0f * PI_F / 35.0f;  // +i for inverse transform
  for (int i = threadIdx.x; i < HWP; i += blockDim.x) {
    int xx = i % HP, ky = i / HP;
    float re = 0.f, im = 0.f;
    for (int kx = 0; kx < HP; ++kx) {
      float s, c;
      __sincosf(w0 * (float)((kx * xx) % HP), &s, &c);
      float ar = aRe[ky * HP + kx], ai = aIm[ky * HP + kx];
      re += ar * c - ai * s;
      im += ar * s + ai * c;
    }
    gRe[i] = re;
    gIm[i] = im;
  }
  __syncthreads();
  const float inv = 1.0f / (float)HWP;
  for (int i = threadIdx.x; i < HWP; i += blockDim.x) {
    int xx = i % HP, yy = i / HP;
    float re = 0.f;
    for (int ky = 0; ky < HP; ++ky) {
      float s, c;
      __sincosf(w0 * (float)((ky * yy) % HP), &s, &c);
      re += gRe[ky * HP + xx] * c - gIm[ky * HP + xx] * s;
    }
    rec[(size_t)img * HWP + i] = re * inv;
  }
}

// ---------------------------------------------------------------------------
// K8: t = wb(1x1) @ c0 + bb + bilinear_upsample(rec, 35->224, half-pixel)
// One thread per (b,y,x); c0 column cached in registers across r.
// ---------------------------------------------------------------------------
__global__ __launch_bounds__(256) void k_bottleneck_up(
    const float* __restrict__ c0, const float* __restrict__ wb,
    const float* __restrict__ bb, const float* __restrict__ rec,
    float* __restrict__ tbuf) {
  int idx = blockIdx.x * blockDim.x + threadIdx.x;
  if (idx >= N1) return;
  int xo = idx % Ww; int t = idx / Ww;
  int yo = t % Hh;   int b = t / Hh;
  const float scale = 35.0f / 224.0f;
  float sy = ((float)yo + 0.5f) * scale - 0.5f;
  float sx = ((float)xo + 0.5f) * scale - 0.5f;
  float y0f = floorf(sy), x0f = floorf(sx);
  float fy = sy - y0f, fx = sx - x0f;
  int y0 = clampi((int)y0f, 0, HP - 1), y1 = clampi((int)y0f + 1, 0, HP - 1);
  int x0 = clampi((int)x0f, 0, HP - 1), x1 = clampi((int)x0f + 1, 0, HP - 1);
  float w00 = (1.f - fy) * (1.f - fx), w01 = (1.f - fy) * fx;
  float w10 = fy * (1.f - fx),         w11 = fy * fx;
  float cv[CR];
  const float* cp = c0 + (size_t)b * CR * HW + yo * Ww + xo;
#pragma unroll
  for (int c = 0; c < CR; ++c) cv[c] = cp[(size_t)c * HW];
  const float* rb = rec + (size_t)b * CR * HWP;
  float* tp = tbuf + (size_t)b * CR * HW + yo * Ww + xo;
  for (int r = 0; r < CR; ++r) {
    float u = bb[r];
    const float* wr = wb + (size_t)r * CR;
#pragma unroll
    for (int c = 0; c < CR; ++c) u = fmaf(wr[c], cv[c], u);
    const float* rc = rb + (size_t)r * HWP;
    float up = w00 * rc[y0 * HP + x0] + w01 * rc[y0 * HP + x1] +
               w10 * rc[y1 * HP + x0] + w11 * rc[y1 * HP + x1];
    tp[(size_t)r * HW] = u + up;
  }
}

// ---------------------------------------------------------------------------
// K9: out = x * sigmoid(w2(1x1) @ t + b2).  GEMM M=256, K=32, N=B*H*W via
// f32 WMMA; 8 waves each own 2 M-tiles of one 16-wide N tile. Fused epilogue.
// Grid = N1/16 = 12544 (exact) -> EXEC all-ones.
// ---------------------------------------------------------------------------
__global__ __launch_bounds__(256) void k_conv1x1_expand_gate(
    const float* __restrict__ tbuf, const float* __restrict__ w2,
    const float* __restrict__ b2, const float* __restrict__ x,
    float* __restrict__ outp) {
  const int lane = threadIdx.x & 31;
  const int wave = threadIdx.x >> 5;
  const int n0   = blockIdx.x * 16;
  const int bIdx = n0 / HW;
  const int hw0  = n0 - bIdx * HW;
  const int col  = lane & 15;
  const int row  = lane & 15;
  const int kk   = (lane < 16) ? 0 : 2;
  const int m0   = wave * 32;

  v8f acc0 = {}, acc1 = {};
  const float* tB  = tbuf + (size_t)bIdx * CR * HW + hw0 + col;
  const float* wA0 = w2 + (size_t)(m0 + row) * CR;
  const float* wA1 = w2 + (size_t)(m0 + 16 + row) * CR;
  for (int k = 0; k < CR; k += 4) {
    v2f bv, a0, a1;
    bv.x = tB[(size_t)(k + kk) * HW];
    bv.y = tB[(size_t)(k + kk + 1) * HW];
    a0.x = wA0[k + kk];
    a0.y = wA0[k + kk + 1];
    a1.x = wA1[k + kk];
    a1.y = wA1[k + kk + 1];
    acc0 = __builtin_amdgcn_wmma_f32_16x16x4_f32(false, a0, false, bv,
                                                 (short)0, acc0, false, false);
    acc1 = __builtin_amdgcn_wmma_f32_16x16x4_f32(false, a1, false, bv,
                                                 (short)0, acc1, false, false);
  }
  const int rowBase = (lane < 16) ? 0 : 8;
#pragma unroll
  for (int half = 0; half < 2; ++half) {
    v8f acc = half ? acc1 : acc0;
    int mt = m0 + half * 16 + rowBase;
#pragma unroll
    for (int r = 0; r < 8; ++r) {
      int o = mt + r;
      size_t oi = ((size_t)bIdx * CIN + o) * HW + hw0 + col;
      float z = acc[r] + b2[o];
      float g = 1.0f / (1.0f + __expf(-z));
      outp[oi] = x[oi] * g;
    }
  }
}

// ---------------------------------------------------------------------------
// Launcher
// ---------------------------------------------------------------------------
extern "C" void kernel_launch(void* const* d_in, const int* in_sizes, int n_in,
                              void* d_out, int out_size, void* d_ws,
                              size_t ws_size, hipStream_t stream) {
  const float* x    = (const float*)d_in[0];
  const float* w1   = (const float*)d_in[1];
  const float* b1   = (const float*)d_in[2];
  const float* wsW  = (const float*)d_in[3];
  const float* bs   = (const float*)d_in[4];
  const float* wb   = (const float*)d_in[5];
  const float* bb   = (const float*)d_in[6];
  const float* woff = (const float*)d_in[7];   // [3,18,32,3,3]
  const float* boff = (const float*)d_in[8];   // [3,18]
  const float* wreg = (const float*)d_in[9];   // [3,32,32,3,3]
  const float* breg = (const float*)d_in[10];  // [3,32]
  const float* w2   = (const float*)d_in[11];
  const float* b2   = (const float*)d_in[12];
  float* out = (float*)d_out;

  // Workspace carve-up (~61 MB of f32; every region written before read).
  float* ws    = (float*)d_ws;
  float* c0    = ws;                                // B*CR*HW
  float* c1    = c0 + (size_t)B_ * CR * HW;         // B*CR*HW1
  float* pbuf  = c1 + (size_t)B_ * CR * HW1;        // B*CR*HWP
  float* hbA   = pbuf + (size_t)B_ * CR * HWP;      // B*CR*HWP
  float* phase = hbA + (size_t)B_ * CR * HWP;       // B*CR*HWP
  float* offb  = phase + (size_t)B_ * CR * HWP;     // B*18*HWP
  float* hbB   = offb + (size_t)B_ * 18 * HWP;      // B*CR*HWP
  float* rec   = hbB + (size_t)B_ * CR * HWP;       // B*CR*HWP
  float* tbuf  = rec + (size_t)B_ * CR * HWP;       // B*CR*HW

  // K1: 1x1 reduce (WMMA f32)
  k_conv1x1_reduce<<<dim3(N1 / 64), dim3(256), 0, stream>>>(x, w1, b1, c0);

  // K2: 3x3 s2 conv
  {
    int total = B_ * CR * HW1;
    k_conv3x3_s2<<<dim3((total + 255) / 256), dim3(256), 0, stream>>>(
        c0, wsW, bs, c1);
  }
  // K3: maxpool 7x7 s3
  {
    int total = B_ * CR * HWP;
    k_maxpool<<<dim3((total + 255) / 256), dim3(256), 0, stream>>>(c1, pbuf);
  }
  // K4: forward DFT + shift + band mask
  k_fft_fwd<<<dim3(B_ * CR), dim3(256), 0, stream>>>(pbuf, hbA, phase);

  // 3 DCN+ReLU blocks (ping-pong hbA <-> hbB)
  float* cur = hbA;
  float* nxt = hbB;
  for (int l = 0; l < 3; ++l) {
    const float* wo = woff + (size_t)l * 18 * CR * 9;
    const float* bo = boff + (size_t)l * 18;
    const float* wr = wreg + (size_t)l * CR * CR * 9;
    const float* br = breg + (size_t)l * CR;
    int totOff = B_ * 18 * HWP;
    k_off_conv<<<dim3((totOff + 255) / 256), dim3(256), 0, stream>>>(
        cur, wo, bo, offb);
    int totDef = B_ * HWP;
    k_deform<<<dim3((totDef + 255) / 256), dim3(256), 0, stream>>>(
        cur, offb, wr, br, nxt);
    float* tmp = cur; cur = nxt; nxt = tmp;
  }
  // after 3 swaps, `cur` holds the final hb

  // K7: inverse DFT -> rec
  k_fft_inv<<<dim3(B_ * CR), dim3(256), 0, stream>>>(cur, phase, rec);

  // K8: t = wb@c0 + bb + upsample(rec)
  k_bottleneck_up<<<dim3((N1 + 255) / 256), dim3(256), 0, stream>>>(
      c0, wb, bb, rec, tbuf);

  // K9: out = x * sigmoid(w2@t + b2)  (WMMA f32, fused gate)
  k_conv1x1_expand_gate<<<dim3(N1 / 16), dim3(256), 0, stream>>>(
      tbuf, w2, b2, x, out);
}